// BioGNN_19241453486765
// MI455X (gfx1250) — compile-verified
//
#include <hip/hip_runtime.h>
#include <math.h>

// ---- problem constants (match reference) ----
#define NN   50000      // nodes
#define EE   800000     // edges (before self-loops)
#define GG   64         // graphs
#define FIN  128        // input features
#define HID  64         // hidden per head
#define NH   4          // heads
#define FOUT 256        // NH*HID
#define NOUT 10
#define ET   (EE + NN)  // edges incl self-loops
#define TILES_M (NN / 16)   // 3125

typedef __attribute__((ext_vector_type(2))) float v2f;
typedef __attribute__((ext_vector_type(8))) float v8f;
typedef int v4i __attribute__((vector_size(16)));   // matches builtin param type

#define AS1 __attribute__((address_space(1)))
#define AS3 __attribute__((address_space(3)))

#if __has_builtin(__builtin_amdgcn_global_load_async_to_lds_b128)
#define HAVE_ASYNC_LDS 1
#else
#define HAVE_ASYNC_LDS 0
#endif

__device__ __forceinline__ void wait_async_lds() {
#if HAVE_ASYNC_LDS
#if __has_builtin(__builtin_amdgcn_s_wait_asynccnt)
    __builtin_amdgcn_s_wait_asynccnt(0);
#else
    asm volatile("s_wait_asynccnt 0" ::: "memory");
#endif
#endif
}

// ---------------------------------------------------------------------------
// fp32 GEMM via native V_WMMA_F32_16X16X4_F32.
// Hout[M, FOUT] = X[M, K] @ W[FOUT, K]^T.
// Block = 8 waves -> 128 rows x 32 cols of output.
//   - W tile (32 rows x K) staged in LDS once per block via async global->LDS
//     (fallback: plain LDS stores). Row stride K+4 floats: conflict-free b64
//     reads, 16B-aligned for B128.
//   - Each wave owns two 16x16 accumulators sharing one A fragment per k-step.
// A fragment: lane(0..15)=row, half-wave selects K pair -> contiguous float2.
// C/D: VGPR q holds M=q (lanes 0-15) / M=q+8 (lanes 16-31), N=lane&15.
// ---------------------------------------------------------------------------
template <int K>
__global__ void gemm_wmma_f32(const float* __restrict__ X,
                              const float* __restrict__ W,
                              float* __restrict__ Hout) {
    static_assert(K % 32 == 0, "K multiple of 32");
    constexpr int LSTR = K + 4;                 // padded LDS row stride (floats)
    __shared__ float sW[32 * LSTR];

    const int w    = threadIdx.x >> 5;          // wave in block (0..7)
    const int lane = threadIdx.x & 31;
    const int tm   = blockIdx.x * 8 + w;        // 16-row tile index
    const int cb   = blockIdx.y * 32;           // output column base
    const bool valid = (tm < TILES_M);
    const int tmc  = valid ? tm : 0;

    // ---- stage W rows [cb, cb+32) into LDS ----
    {
        constexpr int Q4_PER_ROW = K / 4;               // float4s per row
        constexpr int PER_THREAD = (32 * Q4_PER_ROW) / 256;
#pragma unroll
        for (int i = 0; i < PER_THREAD; ++i) {
            int f   = threadIdx.x + 256 * i;            // flat float4 index
            int row = f / Q4_PER_ROW;
            int kq  = f % Q4_PER_ROW;
            const float* gp = W + (size_t)(cb + row) * K + kq * 4;
            float* lp = &sW[row * LSTR + kq * 4];
#if HAVE_ASYNC_LDS
            __builtin_amdgcn_global_load_async_to_lds_b128(
                (AS1 v4i*)gp, (AS3 v4i*)lp, 0, 0);
#else
            *(float4*)lp = *(const float4*)gp;
#endif
        }
        wait_async_lds();
        __syncthreads();
    }

    const int r  = lane & 15;                   // A row within tile / out col
    const int hi = lane >> 4;                   // K-pair selector

    const float* xrow = X + (size_t)(tmc * 16 + r) * K;
    const float* b0p  = &sW[r * LSTR];          // cols cb..cb+15
    const float* b1p  = &sW[(r + 16) * LSTR];   // cols cb+16..cb+31

    v8f acc0 = {}, acc1 = {};
#pragma unroll 8
    for (int k0 = 0; k0 < K; k0 += 4) {
        v2f a  = *(const v2f*)(xrow + k0 + 2 * hi);
        v2f b0 = *(const v2f*)(b0p + k0 + 2 * hi);
        v2f b1 = *(const v2f*)(b1p + k0 + 2 * hi);
        acc0 = __builtin_amdgcn_wmma_f32_16x16x4_f32(false, a, false, b0,
                                                     (short)0, acc0, false, false);
        acc1 = __builtin_amdgcn_wmma_f32_16x16x4_f32(false, a, false, b1,
                                                     (short)0, acc1, false, false);
    }

    if (valid) {
        float* o0 = Hout + (size_t)(tm * 16 + 8 * hi) * FOUT + cb + r;
#pragma unroll
        for (int q = 0; q < 8; ++q) {
            o0[(size_t)q * FOUT]      = acc0[q];
            o0[(size_t)q * FOUT + 16] = acc1[q];
        }
    }
}

// ---------------------------------------------------------------------------
// e_src / e_dst per (node, head):  dot(h[node, head, :], a_src/a_dst[head, :])
// ---------------------------------------------------------------------------
__global__ void attn_coef(const float* __restrict__ H,
                          const float* __restrict__ asrc,
                          const float* __restrict__ adst,
                          float* __restrict__ esrc,
                          float* __restrict__ edst) {
    int idx = blockIdx.x * blockDim.x + threadIdx.x;
    if (idx >= NN * NH) return;
    int node = idx >> 2, head = idx & 3;
    const float* hp = H + (size_t)node * FOUT + head * HID;
    const float* as = asrc + head * HID;
    const float* ad = adst + head * HID;
    float s = 0.f, d = 0.f;
#pragma unroll 8
    for (int c = 0; c < HID; ++c) {
        float h = hp[c];
        s += h * as[c];
        d += h * ad[c];
    }
    esrc[idx] = s;
    edst[idx] = d;
}

// ---------------------------------------------------------------------------
// per-layer scratch init:  m = -inf-ish, denom = 0, agg = 0
// ---------------------------------------------------------------------------
__global__ void layer_init(float* __restrict__ m,
                           float* __restrict__ denom,
                           float* __restrict__ agg) {
    int i = blockIdx.x * blockDim.x + threadIdx.x;
    if (i < NN * NH) { m[i] = -1e30f; denom[i] = 0.f; }
    if (i < NN * FOUT) agg[i] = 0.f;
}

__device__ __forceinline__ float atomicMaxF(float* addr, float val) {
    if (val >= 0.f)
        return __int_as_float(atomicMax((int*)addr, __float_as_int(val)));
    return __uint_as_float(atomicMin((unsigned int*)addr,
                                     (unsigned int)__float_as_int(val)));
}

__device__ __forceinline__ void edge_nodes(int e, const int* src, const int* dst,
                                           int& s, int& d) {
    if (e < EE) { s = src[e]; d = dst[e]; }
    else        { s = d = e - EE; }        // self-loops appended
}

// ---------------------------------------------------------------------------
// segment max of leaky_relu(e_src[s] + e_dst[d]) over dst
// ---------------------------------------------------------------------------
__global__ void edge_max(const int* __restrict__ src, const int* __restrict__ dst,
                         const float* __restrict__ esrc,
                         const float* __restrict__ edst,
                         float* __restrict__ m) {
    int e = blockIdx.x * blockDim.x + threadIdx.x;
    if (e >= ET) return;
    int s, d;
    edge_nodes(e, src, dst, s, d);
#pragma unroll
    for (int h = 0; h < NH; ++h) {
        float l = esrc[s * NH + h] + edst[d * NH + h];
        l = (l > 0.f) ? l : 0.2f * l;
        atomicMaxF(&m[d * NH + h], l);
    }
}

// ---------------------------------------------------------------------------
// ex = exp(logit - m[dst]); store per-edge numerator, atomic-add denominator
// ---------------------------------------------------------------------------
__global__ void edge_expsum(const int* __restrict__ src, const int* __restrict__ dst,
                            const float* __restrict__ esrc,
                            const float* __restrict__ edst,
                            const float* __restrict__ m,
                            float* __restrict__ exbuf,
                            float* __restrict__ denom) {
    int e = blockIdx.x * blockDim.x + threadIdx.x;
    if (e >= ET) return;
    int s, d;
    edge_nodes(e, src, dst, s, d);
#pragma unroll
    for (int h = 0; h < NH; ++h) {
        float l = esrc[s * NH + h] + edst[d * NH + h];
        l = (l > 0.f) ? l : 0.2f * l;
        float ex = __expf(l - m[d * NH + h]);
        exbuf[(size_t)e * NH + h] = ex;
        atomicAdd(&denom[d * NH + h], ex);
    }
}

// ---------------------------------------------------------------------------
// message passing: one wave32 per edge, 8 channels per lane.
// agg[dst, ch] += h[src, ch] * alpha[head(ch)]
// ---------------------------------------------------------------------------
__global__ void edge_msg(const int* __restrict__ src, const int* __restrict__ dst,
                         const float* __restrict__ H,
                         const float* __restrict__ exbuf,
                         const float* __restrict__ denom,
                         float* __restrict__ agg) {
    int wave = (blockIdx.x * blockDim.x + threadIdx.x) >> 5;
    int lane = threadIdx.x & 31;
    if (wave >= ET) return;
    int s, d;
    edge_nodes(wave, src, dst, s, d);
    float alpha[NH];
#pragma unroll
    for (int h = 0; h < NH; ++h)
        alpha[h] = exbuf[(size_t)wave * NH + h] / (denom[d * NH + h] + 1e-16f);
    const float* hs = H + (size_t)s * FOUT;
    float* ag = agg + (size_t)d * FOUT;
#pragma unroll
    for (int j = 0; j < FOUT / 32; ++j) {
        int ch = lane + 32 * j;
        atomicAdd(&ag[ch], hs[ch] * alpha[ch >> 6]);   // HID=64 -> head = ch>>6
    }
}

// ---------------------------------------------------------------------------
// out = (agg + b), optional ELU
// ---------------------------------------------------------------------------
__global__ void bias_act(const float* __restrict__ agg,
                         const float* __restrict__ b,
                         float* __restrict__ out, int do_elu) {
    long long i = (long long)blockIdx.x * blockDim.x + threadIdx.x;
    if (i >= (long long)NN * FOUT) return;
    float v = agg[i] + b[(int)(i & (FOUT - 1))];
    if (do_elu) v = (v > 0.f) ? v : (__expf(v) - 1.f);
    out[i] = v;
}

// ---------------------------------------------------------------------------
// pooling + final linear
// ---------------------------------------------------------------------------
__global__ void pool_init(float* __restrict__ pool, float* __restrict__ cnt) {
    int i = blockIdx.x * blockDim.x + threadIdx.x;
    if (i < GG * FOUT) pool[i] = 0.f;
    if (i < GG) cnt[i] = 0.f;
}

__global__ void pool_sum(const float* __restrict__ X,
                         const int* __restrict__ batch,
                         float* __restrict__ pool, float* __restrict__ cnt) {
    long long i = (long long)blockIdx.x * blockDim.x + threadIdx.x;
    if (i >= (long long)NN * FOUT) return;
    int node = (int)(i >> 8);
    int ch   = (int)(i & 255);
    int g = batch[node];
    atomicAdd(&pool[g * FOUT + ch], X[i]);
    if (ch == 0) atomicAdd(&cnt[g], 1.f);
}

__global__ void final_linear(const float* __restrict__ pool,
                             const float* __restrict__ cnt,
                             const float* __restrict__ Wl,
                             const float* __restrict__ bl,
                             float* __restrict__ out) {
    int i = blockIdx.x * blockDim.x + threadIdx.x;
    if (i >= GG * NOUT) return;
    int g = i / NOUT, o = i % NOUT;
    float inv = 1.f / fmaxf(cnt[g], 1.f);
    float acc = bl[o];
#pragma unroll 8
    for (int k = 0; k < FOUT; ++k)
        acc += pool[g * FOUT + k] * inv * Wl[o * FOUT + k];
    out[i] = acc;
}

// ---------------------------------------------------------------------------
// host side
// ---------------------------------------------------------------------------
template <int K>
static void run_gat_layer(const float* Xin,
                          const float* W, const float* asrc, const float* adst,
                          const float* b,
                          const int* src, const int* dst,
                          float* Hb, float* Agg, float* Xout,
                          float* esrc, float* edst, float* m, float* denom,
                          float* exbuf, int do_elu, hipStream_t stream) {
    const int BLK = 256;
    // GEMM: 8 waves/block, 128x32 output per block
    {
        dim3 grid((TILES_M + 7) / 8, FOUT / 32);
        gemm_wmma_f32<K><<<grid, BLK, 0, stream>>>(Xin, W, Hb);
    }
    attn_coef<<<(NN * NH + BLK - 1) / BLK, BLK, 0, stream>>>(Hb, asrc, adst, esrc, edst);
    layer_init<<<(NN * FOUT + BLK - 1) / BLK, BLK, 0, stream>>>(m, denom, Agg);
    edge_max<<<(ET + BLK - 1) / BLK, BLK, 0, stream>>>(src, dst, esrc, edst, m);
    edge_expsum<<<(ET + BLK - 1) / BLK, BLK, 0, stream>>>(src, dst, esrc, edst, m,
                                                          exbuf, denom);
    {
        long long thr = (long long)ET * 32;
        int blocks = (int)((thr + BLK - 1) / BLK);
        edge_msg<<<blocks, BLK, 0, stream>>>(src, dst, Hb, exbuf, denom, Agg);
    }
    bias_act<<<(NN * FOUT + BLK - 1) / BLK, BLK, 0, stream>>>(Agg, b, Xout, do_elu);
}

extern "C" void kernel_launch(void* const* d_in, const int* in_sizes, int n_in,
                              void* d_out, int out_size, void* d_ws, size_t ws_size,
                              hipStream_t stream) {
    // inputs (setup_inputs order)
    const float* x     = (const float*)d_in[0];
    const int*   ei    = (const int*)d_in[1];     // [2, E]
    const int*   batch = (const int*)d_in[2];
    const float* W1 = (const float*)d_in[3];
    const float* a_src1 = (const float*)d_in[4];
    const float* a_dst1 = (const float*)d_in[5];
    const float* b1 = (const float*)d_in[6];
    const float* W2 = (const float*)d_in[7];
    const float* a_src2 = (const float*)d_in[8];
    const float* a_dst2 = (const float*)d_in[9];
    const float* b2 = (const float*)d_in[10];
    const float* W3 = (const float*)d_in[11];
    const float* a_src3 = (const float*)d_in[12];
    const float* a_dst3 = (const float*)d_in[13];
    const float* b3 = (const float*)d_in[14];
    const float* Wl = (const float*)d_in[15];
    const float* bl = (const float*)d_in[16];

    const int* src = ei;          // row 0
    const int* dst = ei + EE;     // row 1

    // workspace partition (floats)
    float* ws = (float*)d_ws;
    size_t off = 0;
    float* FB0   = ws + off; off += (size_t)NN * FOUT;   // activated features
    float* FB1   = ws + off; off += (size_t)NN * FOUT;
    float* Hb    = ws + off; off += (size_t)NN * FOUT;   // GEMM output
    float* Agg   = ws + off; off += (size_t)NN * FOUT;   // aggregation
    float* esrc  = ws + off; off += (size_t)NN * NH;
    float* edstb = ws + off; off += (size_t)NN * NH;
    float* m     = ws + off; off += (size_t)NN * NH;
    float* denom = ws + off; off += (size_t)NN * NH;
    float* exbuf = ws + off; off += (size_t)ET * NH;
    float* pool  = ws + off; off += (size_t)GG * FOUT;
    float* cnt   = ws + off; off += (size_t)GG;
    (void)ws_size; (void)n_in; (void)in_sizes; (void)out_size;

    // layer 1: x [N,128] -> FB0
    run_gat_layer<FIN>(x, W1, a_src1, a_dst1, b1, src, dst,
                       Hb, Agg, FB0, esrc, edstb, m, denom, exbuf, 1, stream);
    // layer 2: FB0 -> FB1
    run_gat_layer<FOUT>(FB0, W2, a_src2, a_dst2, b2, src, dst,
                        Hb, Agg, FB1, esrc, edstb, m, denom, exbuf, 1, stream);
    // layer 3: FB1 -> FB0 (no ELU)
    run_gat_layer<FOUT>(FB1, W3, a_src3, a_dst3, b3, src, dst,
                        Hb, Agg, FB0, esrc, edstb, m, denom, exbuf, 0, stream);

    // global mean pool + linear
    const int BLK = 256;
    pool_init<<<(GG * FOUT + BLK - 1) / BLK, BLK, 0, stream>>>(pool, cnt);
    pool_sum<<<(NN * FOUT + BLK - 1) / BLK, BLK, 0, stream>>>(FB0, batch, pool, cnt);
    final_linear<<<(GG * NOUT + BLK - 1) / BLK, BLK, 0, stream>>>(pool, cnt, Wl, bl,
                                                                  (float*)d_out);
}